// EncLayer_3564822855799
// MI455X (gfx1250) — compile-verified
//
#include <hip/hip_runtime.h>
#include <math.h>

typedef __attribute__((ext_vector_type(2))) float v2f;
typedef __attribute__((ext_vector_type(8))) float v8f;

#define B_  2
#define N_  2048
#define K_  48
#define C_  128
#define FF_ 512
#define LDE 132   // padded LDS row stride (132 % 64 == 4 -> 16 rows hit distinct banks)

__device__ __forceinline__ v8f wmma4(v2f a, v2f b, v8f c) {
  // V_WMMA_F32_16X16X4_F32 : D = A(16x4) * B(4x16) + C(16x16), full fp32
  return __builtin_amdgcn_wmma_f32_16x16x4_f32(false, a, false, b, (short)0, c, false, false);
}

__device__ __forceinline__ float gelu_f(float x) {
  return 0.5f * x * (1.0f + erff(x * 0.7071067811865475f));  // exact GELU
}

// ---------------------------------------------------------------------------
// Repack W[Kd][Nd] into WMMA B-fragment order:
//   Wp[(nt*(Kd/4) + k4)*64 + lane*2 + v] = W[k4*4 + 2*(lane>>4) + v][nt*16 + (lane&15)]
// so each lane fetches its fragment pair with ONE coalesced global_load_b64.
// ---------------------------------------------------------------------------
__global__ void __launch_bounds__(256)
repack_w(const float* __restrict__ W, float* __restrict__ Wp, int Kd, int Nd)
{
  int gid = blockIdx.x * 256 + threadIdx.x;         // one per lane-pair
  if (gid >= (Kd * Nd) / 2) return;
  int lane = gid & 31;
  int tile = gid >> 5;                              // nt*(Kd/4) + k4
  int kt   = Kd >> 2;
  int k4   = tile % kt;
  int nt   = tile / kt;
  int krow = k4 * 4 + 2 * (lane >> 4);
  int col  = nt * 16 + (lane & 15);
  Wp[(size_t)gid * 2 + 0] = W[(size_t)krow * Nd + col];
  Wp[(size_t)gid * 2 + 1] = W[(size_t)(krow + 1) * Nd + col];
}

// ---------------------------------------------------------------------------
// Dense GEMM: Y[M,N] = act(X[M,K] @ W[K,N] + bias[N]), W pre-packed.
// One 16-row strip per block (256 thr = 8 waves); X strip staged in LDS.
// ---------------------------------------------------------------------------
template<int KDIM, int NDIM, bool ACT>
__global__ void __launch_bounds__(256)
gemm_bias_act(const float* __restrict__ X, const float* __restrict__ Wp,
              const float* __restrict__ bias, float* __restrict__ Y)
{
  constexpr int LDX = KDIM + 4;           // (KDIM+4)%64==4 -> conflict-free
  constexpr int KT  = KDIM / 4;
  extern __shared__ float smem[];
  const int m0  = blockIdx.x * 16;
  const int tid = threadIdx.x;

  for (int i = tid * 4; i < 16 * KDIM; i += 1024) {
    int row = i / KDIM, k = i % KDIM;
    *(float4*)(smem + row * LDX + k) =
        *(const float4*)(X + (size_t)(m0 + row) * KDIM + k);
  }
  __syncthreads();

  const int wave = tid >> 5, lane = tid & 31, half = lane >> 4, l16 = lane & 15;

  for (int nt = wave; nt < NDIM / 16; nt += 8) {
    const int n0 = nt * 16;
    float bv = bias[n0 + l16];
    v8f acc;
    for (int r = 0; r < 8; ++r) acc[r] = bv;

    const float* wb = Wp + (size_t)nt * KT * 64 + lane * 2;
    for (int k4 = 0; k4 < KT; ++k4) {
      const float* ap = smem + l16 * LDX + k4 * 4 + 2 * half;
      v2f a; a[0] = ap[0]; a[1] = ap[1];
      v2f b = *(const v2f*)(wb + k4 * 64);
      acc = wmma4(a, b, acc);
    }
    for (int r = 0; r < 8; ++r) {
      float v = acc[r];
      if (ACT) v = gelu_f(v);
      Y[(size_t)(m0 + r + 8 * half) * NDIM + n0 + l16] = v;
    }
  }
}

// 6-tile (3 M-tiles x 2 N-tiles) K=128 GEMM step for the edge-MLP layers.
// Wp is a packed 128x128 weight (8 n-tiles, 32 k-steps).
__device__ __forceinline__ void gemm_tile6(v8f acc[3][2], const float* Abuf,
                                           const float* __restrict__ Wp,
                                           int wave, int lane, int half, int l16)
{
  const float* wb0 = Wp + (size_t)(wave * 2 + 0) * 32 * 64 + lane * 2;
  const float* wb1 = Wp + (size_t)(wave * 2 + 1) * 32 * 64 + lane * 2;
  for (int k4 = 0; k4 < 32; ++k4) {
    v2f a[3];
    for (int m = 0; m < 3; ++m) {
      const float* p = Abuf + (16 * m + l16) * LDE + k4 * 4 + 2 * half;
      a[m][0] = p[0]; a[m][1] = p[1];
    }
    v2f b0 = *(const v2f*)(wb0 + k4 * 64);
    v2f b1 = *(const v2f*)(wb1 + k4 * 64);
    for (int m = 0; m < 3; ++m) acc[m][0] = wmma4(a[m], b0, acc[m][0]);
    for (int m = 0; m < 3; ++m) acc[m][1] = wmma4(a[m], b1, acc[m][1]);
  }
}

// ---------------------------------------------------------------------------
// Fused per-node edge MLP.  One block (4 waves) per node.
// MODE 0: dh[node,c] = (1/30) * sum_k mask * MLP([ctr,hE,nbr]); ctr term
//         pre-folded into bias1.  h2 overlays eBuf (h_E dead after layer 1).
// MODE 1: out[node,k,c] = LN(h_E + MLP(...); g3,b3)
// ---------------------------------------------------------------------------
template<int MODE>
__global__ void __launch_bounds__(128)
edge_mlp_kernel(const float* __restrict__ hsrc, const float* __restrict__ hE,
                const int* __restrict__ Eidx, const float* __restrict__ maskA,
                const float* __restrict__ bias1, const float* __restrict__ pW1b,
                const float* __restrict__ pW1c,
                const float* __restrict__ pW2, const float* __restrict__ W2b,
                const float* __restrict__ pW3, const float* __restrict__ W3b,
                const float* __restrict__ g3, const float* __restrict__ b3ln,
                float* __restrict__ out)
{
  extern __shared__ float smem[];
  float* eBuf = smem;                                    // [K_][LDE]
  float* nBuf = eBuf + K_ * LDE;                         // gather; later h3 (mode 1)
  float* h1B  = nBuf + K_ * LDE;
  float* h2B  = (MODE == 0) ? eBuf : h1B + K_ * LDE;     // overlay in mode 0
  float* mBuf = h1B + K_ * LDE;                          // mode 0 only
  float* dhB  = mBuf + K_;                               // mode 0 only

  const int node = blockIdx.x;
  const int tid  = threadIdx.x;
  const int wave = tid >> 5, lane = tid & 31, half = lane >> 4, l16 = lane & 15;

  const float* eSrc     = hE + (size_t)node * (K_ * C_);
  const int*   idxp     = Eidx + (size_t)node * K_;
  const float* nodeBase = hsrc + (size_t)(node >> 11) * ((size_t)N_ * C_);

  for (int i = tid; i < K_ * (C_ / 4); i += 128) {
    int row = i >> 5;
    int c   = (i & 31) << 2;
    *(float4*)(eBuf + row * LDE + c) = *(const float4*)(eSrc + row * C_ + c);
    int nb = idxp[row];
    *(float4*)(nBuf + row * LDE + c) =
        *(const float4*)(nodeBase + (size_t)nb * C_ + c);
  }
  if (MODE == 0) {
    if (tid < K_) mBuf[tid] = maskA[(size_t)node * K_ + tid];
    if (tid < C_) dhB[tid] = 0.0f;
  }
  __syncthreads();

  const int col0 = wave * 32 + l16;   // +16 for the second n-tile
  v8f acc[3][2];

  // ---- layer 1: h1 = gelu(hE@W1b + nbr@W1c + bias1) ----
  for (int t = 0; t < 2; ++t) {
    float bv = bias1[(size_t)node * C_ + col0 + t * 16];
    for (int m = 0; m < 3; ++m)
      for (int r = 0; r < 8; ++r) acc[m][t][r] = bv;
  }
  gemm_tile6(acc, eBuf, pW1b, wave, lane, half, l16);
  gemm_tile6(acc, nBuf, pW1c, wave, lane, half, l16);
  __syncthreads();          // all eBuf/nBuf reads done before h1 (and later h2) writes
  for (int m = 0; m < 3; ++m)
    for (int t = 0; t < 2; ++t)
      for (int r = 0; r < 8; ++r)
        h1B[(16 * m + r + 8 * half) * LDE + col0 + t * 16] = gelu_f(acc[m][t][r]);
  __syncthreads();

  // ---- layer 2: h2 = gelu(h1@W2 + b2) ----
  for (int t = 0; t < 2; ++t) {
    float bv = W2b[col0 + t * 16];
    for (int m = 0; m < 3; ++m)
      for (int r = 0; r < 8; ++r) acc[m][t][r] = bv;
  }
  gemm_tile6(acc, h1B, pW2, wave, lane, half, l16);
  for (int m = 0; m < 3; ++m)
    for (int t = 0; t < 2; ++t)
      for (int r = 0; r < 8; ++r)
        h2B[(16 * m + r + 8 * half) * LDE + col0 + t * 16] = gelu_f(acc[m][t][r]);
  __syncthreads();

  // ---- layer 3: h3 = h2@W3 + b3 ----
  for (int t = 0; t < 2; ++t) {
    float bv = W3b[col0 + t * 16];
    for (int m = 0; m < 3; ++m)
      for (int r = 0; r < 8; ++r) acc[m][t][r] = bv;
  }
  gemm_tile6(acc, h2B, pW3, wave, lane, half, l16);

  if (MODE == 0) {
    // masked column sums over the 48 edges -> dh
    for (int t = 0; t < 2; ++t) {
      float cs = 0.0f;
      for (int m = 0; m < 3; ++m)
        for (int r = 0; r < 8; ++r)
          cs += acc[m][t][r] * mBuf[16 * m + r + 8 * half];
      cs += __shfl_xor(cs, 16);                 // lanes l and l^16 share column
      if (half == 0) atomicAdd(&dhB[col0 + t * 16], cs);
    }
    __syncthreads();
    if (tid < C_) out[(size_t)node * C_ + tid] = dhB[tid] * (1.0f / 30.0f);
  } else {
    // stage h3 (overlay nBuf), then per-edge residual LayerNorm against h_E
    for (int m = 0; m < 3; ++m)
      for (int t = 0; t < 2; ++t)
        for (int r = 0; r < 8; ++r)
          nBuf[(16 * m + r + 8 * half) * LDE + col0 + t * 16] = acc[m][t][r];
    __syncthreads();
    for (int row = wave; row < K_; row += 4) {
      float s = 0.0f, ss = 0.0f, vals[4];
      for (int j = 0; j < 4; ++j) {
        int c = lane + 32 * j;
        float x = eBuf[row * LDE + c] + nBuf[row * LDE + c];
        vals[j] = x; s += x; ss += x * x;
      }
      for (int o = 16; o > 0; o >>= 1) {
        s += __shfl_xor(s, o); ss += __shfl_xor(ss, o);
      }
      float mean = s * (1.0f / C_);
      float inv  = rsqrtf(ss * (1.0f / C_) - mean * mean + 1e-5f);
      for (int j = 0; j < 4; ++j) {
        int c = lane + 32 * j;
        out[((size_t)node * K_ + row) * C_ + c] =
            g3[c] * (vals[j] - mean) * inv + b3ln[c];
      }
    }
  }
}

// ---------------------------------------------------------------------------
// Residual LayerNorm over C=128, one wave per node, shfl reductions.
// out = mask * ( g*(x+d - mean)*rsqrt(var+eps) + b )
// ---------------------------------------------------------------------------
__global__ void __launch_bounds__(128)
ln_kernel(const float* __restrict__ x, const float* __restrict__ d,
          const float* __restrict__ g, const float* __restrict__ bt,
          const float* __restrict__ maskv, float* __restrict__ out, int nNodes)
{
  int node = blockIdx.x * 4 + (threadIdx.x >> 5);
  if (node >= nNodes) return;
  int lane = threadIdx.x & 31;
  float vals[4], s = 0.0f, ss = 0.0f;
  for (int j = 0; j < 4; ++j) {
    int c = lane + 32 * j;
    float v = x[(size_t)node * C_ + c] + d[(size_t)node * C_ + c];
    vals[j] = v; s += v; ss += v * v;
  }
  for (int o = 16; o > 0; o >>= 1) { s += __shfl_xor(s, o); ss += __shfl_xor(ss, o); }
  float mean = s * (1.0f / C_);
  float inv  = rsqrtf(ss * (1.0f / C_) - mean * mean + 1e-5f);
  float mk   = maskv ? maskv[node] : 1.0f;
  for (int j = 0; j < 4; ++j) {
    int c = lane + 32 * j;
    out[(size_t)node * C_ + c] = mk * (g[c] * (vals[j] - mean) * inv + bt[c]);
  }
}

// ---------------------------------------------------------------------------
extern "C" void kernel_launch(void* const* d_in, const int* in_sizes, int n_in,
                              void* d_out, int out_size, void* d_ws, size_t ws_size,
                              hipStream_t stream)
{
  const float* h_V         = (const float*)d_in[0];
  const float* h_E         = (const float*)d_in[1];
  const int*   E_idx       = (const int*)  d_in[2];
  const float* mask_V      = (const float*)d_in[3];
  const float* mask_attend = (const float*)d_in[4];
  const float* W1_w  = (const float*)d_in[5];  const float* W1_b  = (const float*)d_in[6];
  const float* W2_w  = (const float*)d_in[7];  const float* W2_b  = (const float*)d_in[8];
  const float* W3_w  = (const float*)d_in[9];  const float* W3_b  = (const float*)d_in[10];
  const float* W11_w = (const float*)d_in[11]; const float* W11_b = (const float*)d_in[12];
  const float* W12_w = (const float*)d_in[13]; const float* W12_b = (const float*)d_in[14];
  const float* W13_w = (const float*)d_in[15]; const float* W13_b = (const float*)d_in[16];
  const float* Win_w = (const float*)d_in[17]; const float* Win_b = (const float*)d_in[18];
  const float* Wout_w= (const float*)d_in[19]; const float* Wout_b= (const float*)d_in[20];
  const float* g1 = (const float*)d_in[21]; const float* b1 = (const float*)d_in[22];
  const float* g2 = (const float*)d_in[23]; const float* b2 = (const float*)d_in[24];
  const float* g3 = (const float*)d_in[25]; const float* b3 = (const float*)d_in[26];

  const int M = B_ * N_;                       // 4096 nodes
  float* ws    = (float*)d_ws;
  float* bias1 = ws;                           // [M,C]
  float* dh    = bias1 + (size_t)M * C_;       // [M,C]
  float* hv1   = dh    + (size_t)M * C_;       // [M,C]
  float* f1    = hv1   + (size_t)M * C_;       // [M,FF]
  float* f2    = f1    + (size_t)M * FF_;      // [M,C]
  float* bias2 = f2    + (size_t)M * C_;       // [M,C]
  // packed weights (fragment order)
  float* p     = bias2 + (size_t)M * C_;
  float* pW1a  = p;           p += 16384;      // 128x128 blocks
  float* pW1b  = p;           p += 16384;
  float* pW1c  = p;           p += 16384;
  float* pW2   = p;           p += 16384;
  float* pW3   = p;           p += 16384;
  float* pW11a = p;           p += 16384;
  float* pW11b = p;           p += 16384;
  float* pW11c = p;           p += 16384;
  float* pW12  = p;           p += 16384;
  float* pW13  = p;           p += 16384;
  float* pWin  = p;           p += 65536;      // 128x512
  float* pWout = p;           p += 65536;      // 512x128

  float* hv2   = (float*)d_out;                // output 0: [B,N,C]
  float* hEout = hv2 + (size_t)M * C_;         // output 1: [B,N,K,C]

  // --- repack all weights into WMMA fragment order (one-time, ~1.1 MB) ---
  repack_w<<<32, 256, 0, stream>>>(W1_w,             pW1a, 128, 128);
  repack_w<<<32, 256, 0, stream>>>(W1_w + 128*128,   pW1b, 128, 128);
  repack_w<<<32, 256, 0, stream>>>(W1_w + 256*128,   pW1c, 128, 128);
  repack_w<<<32, 256, 0, stream>>>(W2_w,             pW2,  128, 128);
  repack_w<<<32, 256, 0, stream>>>(W3_w,             pW3,  128, 128);
  repack_w<<<32, 256, 0, stream>>>(W11_w,            pW11a,128, 128);
  repack_w<<<32, 256, 0, stream>>>(W11_w + 128*128,  pW11b,128, 128);
  repack_w<<<32, 256, 0, stream>>>(W11_w + 256*128,  pW11c,128, 128);
  repack_w<<<32, 256, 0, stream>>>(W12_w,            pW12, 128, 128);
  repack_w<<<32, 256, 0, stream>>>(W13_w,            pW13, 128, 128);
  repack_w<<<128,256, 0, stream>>>(Win_w,            pWin, 128, 512);
  repack_w<<<128,256, 0, stream>>>(Wout_w,           pWout,512, 128);

  const size_t smem0 = (size_t)(3 * K_ * LDE + K_ + C_) * sizeof(float); // 76.7 KB
  const size_t smem1 = (size_t)(4 * K_ * LDE) * sizeof(float);           // 101.4 KB

  // 1. ctr term of edge-MLP-1 folded into per-node bias
  gemm_bias_act<128, 128, false><<<M / 16, 256, 16 * 132 * sizeof(float), stream>>>(
      h_V, pW1a, W1_b, bias1);
  // 2. per-node fused edge MLP -> dh (already / SCALE)
  edge_mlp_kernel<0><<<M, 128, smem0, stream>>>(
      h_V, h_E, E_idx, mask_attend, bias1, pW1b, pW1c, pW2, W2_b, pW3, W3_b,
      nullptr, nullptr, dh);
  // 3. hv1 = LN(h_V + dh)
  ln_kernel<<<M / 4, 128, 0, stream>>>(h_V, dh, g1, b1, nullptr, hv1, M);
  // 4. f1 = gelu(hv1 @ Win + b)
  gemm_bias_act<128, 512, true><<<M / 16, 256, 16 * 132 * sizeof(float), stream>>>(
      hv1, pWin, Win_b, f1);
  // 5. f2 = f1 @ Wout + b
  gemm_bias_act<512, 128, false><<<M / 16, 256, 16 * 516 * sizeof(float), stream>>>(
      f1, pWout, Wout_b, f2);
  // 6. hv2 = mask_V * LN(hv1 + f2)
  ln_kernel<<<M / 4, 128, 0, stream>>>(hv1, f2, g2, b2, mask_V, hv2, M);
  // 7. ctr term of edge-MLP-2
  gemm_bias_act<128, 128, false><<<M / 16, 256, 16 * 132 * sizeof(float), stream>>>(
      hv2, pW11a, W11_b, bias2);
  // 8. edge update + per-edge residual LN -> h_E out
  edge_mlp_kernel<1><<<M, 128, smem1, stream>>>(
      hv2, h_E, E_idx, nullptr, bias2, pW11b, pW11c, pW12, W12_b, pW13, W13_b,
      g3, b3, hEout);
}